// ValueNorm_90340342104515
// MI455X (gfx1250) — compile-verified
//
#include <hip/hip_runtime.h>

// ValueNorm (Welford merge + normalize) for MI455X / gfx1250, wave32.
// K1: streaming partial reduction (sum, sumsq) -> ws
// K2: WMMA-based final reduction tree + Chan merge scalars
// K3: streaming normalize with NT stores

typedef float v2f __attribute__((ext_vector_type(2)));
typedef float v4f __attribute__((ext_vector_type(4)));
typedef float v8f __attribute__((ext_vector_type(8)));

#define EPSV 1e-5f
#define RED_BLOCKS 2048
#define NRM_BLOCKS 4096
#define TPB 256

// D = A(16x4) x B(4x16) + C, fp32 exact
__device__ __forceinline__ v8f wmma4(v2f a, v2f b, v8f c) {
  return __builtin_amdgcn_wmma_f32_16x16x4_f32(
      /*neg_a=*/false, a, /*neg_b=*/false, b,
      /*c_mod=*/(short)0, c, /*reuse_a=*/false, /*reuse_b=*/false);
}

// Reduce 256 floats in LDS to one scalar using the matrix unit.
// Stage 1: 4 accumulating WMMAs with B=ones -> 16 row-group sums R_m
//          (D[m][n] == R_m for all n).
// Stage 2: D's VGPR pairs are already in B-operand layout
//          (B[0]/B[1] from lanes 0-15 == D[2j]/D[2j+1], B[2]/B[3] from
//          lanes 16-31 == D[2j+8]/D[2j+9]), so 4 accumulating WMMAs with
//          A=ones sum all 16 R_m -> total replicated in every D element.
__device__ __forceinline__ float wmma_reduce256(const float* sh) {
  const int lane = (int)(threadIdx.x & 31u);
  // A-layout: lane L<16 holds A[L][0..1]; lane L>=16 holds A[L-16][2..3]
  const int idx = (lane & 15) * 4 + ((lane & 16) ? 2 : 0);
  v2f ones; ones[0] = 1.0f; ones[1] = 1.0f;
  v8f acc = {};
#pragma unroll
  for (int j = 0; j < 4; ++j) {
    v2f a;
    a[0] = sh[j * 64 + idx];
    a[1] = sh[j * 64 + idx + 1];
    acc = wmma4(a, ones, acc);
  }
  v8f tot = {};
#pragma unroll
  for (int j = 0; j < 4; ++j) {
    v2f b;
    b[0] = acc[2 * j];
    b[1] = acc[2 * j + 1];
    tot = wmma4(ones, b, tot);
  }
  return tot[0];
}

__global__ void __launch_bounds__(TPB)
vn_partial_reduce(const v4f* __restrict__ x4, int n4,
                  float* __restrict__ ps, float* __restrict__ pq) {
  const int tid = (int)threadIdx.x;
  const int stride = (int)(gridDim.x * blockDim.x);
  float s = 0.0f, q = 0.0f;
  for (int i = (int)(blockIdx.x * blockDim.x) + tid; i < n4; i += stride) {
    __builtin_prefetch((const char*)&x4[i] + 4096, 0, 1);
    v4f v = x4[i];
    s += (v[0] + v[1]) + (v[2] + v[3]);
    q = fmaf(v[0], v[0], q);
    q = fmaf(v[1], v[1], q);
    q = fmaf(v[2], v[2], q);
    q = fmaf(v[3], v[3], q);
  }
  __shared__ float shs[TPB];
  __shared__ float shq[TPB];
  shs[tid] = s;
  shq[tid] = q;
  __syncthreads();
  // deterministic fixed-order tree (no float atomics -> replay-stable)
#pragma unroll
  for (int off = TPB / 2; off > 0; off >>= 1) {
    if (tid < off) {
      shs[tid] += shs[tid + off];
      shq[tid] += shq[tid + off];
    }
    __syncthreads();
  }
  if (tid == 0) {
    ps[blockIdx.x] = shs[0];
    pq[blockIdx.x] = shq[0];
  }
}

__global__ void __launch_bounds__(TPB)
vn_finalize(const float* __restrict__ ps, const float* __restrict__ pq,
            const float* __restrict__ count_in, const float* __restrict__ mean_in,
            const float* __restrict__ m2_in,
            float* __restrict__ out_scalars,   // d_out + N : {count, mean, m2}
            float* __restrict__ ws_scalars,    // {new_mean, inv_std}
            int nblocks, float Nf) {
  __shared__ float sh[2 * TPB];
  const int t = (int)threadIdx.x;
  float s = 0.0f, q = 0.0f;
  for (int i = t; i < nblocks; i += TPB) {
    s += ps[i];
    q += pq[i];
  }
  sh[t] = s;
  sh[TPB + t] = q;
  __syncthreads();
  // Every wave runs the WMMA tree redundantly -> EXEC stays all-1s for WMMA.
  float S = wmma_reduce256(sh);
  float Q = wmma_reduce256(sh + TPB);
  if (t == 0) {
    const float count = count_in[0];
    const float mean  = mean_in[0];
    const float m2    = m2_in[0];
    const float bmean = S / Nf;
    const float bm2   = fmaf(-Nf * bmean, bmean, Q);   // sum((x-bmean)^2)
    const float new_count = count + Nf;
    const float delta     = bmean - mean;
    const float new_mean  = mean + delta * Nf / new_count;
    const float new_m2    = m2 + bm2 + delta * delta * count * Nf / new_count;
    const float var       = new_m2 / fmaxf(new_count - 1.0f, 1.0f);
    const float inv_std   = 1.0f / sqrtf(var + EPSV);
    out_scalars[0] = new_count;
    out_scalars[1] = new_mean;
    out_scalars[2] = new_m2;
    ws_scalars[0]  = new_mean;
    ws_scalars[1]  = inv_std;
  }
}

__global__ void __launch_bounds__(TPB)
vn_normalize(const v4f* __restrict__ x4, v4f* __restrict__ y4,
             const float* __restrict__ ws_scalars, int n4) {
  const float mean = ws_scalars[0];
  const float inv  = ws_scalars[1];
  const float bias = -mean * inv;           // y = fma(x, inv, bias)
  const int stride = (int)(gridDim.x * blockDim.x);
  for (int i = (int)(blockIdx.x * blockDim.x) + (int)threadIdx.x; i < n4;
       i += stride) {
    __builtin_prefetch((const char*)&x4[i] + 4096, 0, 1);
    v4f v = x4[i];
    v4f r;
    r[0] = fmaf(v[0], inv, bias);
    r[1] = fmaf(v[1], inv, bias);
    r[2] = fmaf(v[2], inv, bias);
    r[3] = fmaf(v[3], inv, bias);
    // y is never re-read: NT store keeps L2 for x's second pass
    __builtin_nontemporal_store(r, &y4[i]);
  }
}

extern "C" void kernel_launch(void* const* d_in, const int* in_sizes, int n_in,
                              void* d_out, int out_size, void* d_ws, size_t ws_size,
                              hipStream_t stream) {
  (void)n_in; (void)out_size; (void)ws_size;
  const float* x        = (const float*)d_in[0];
  const float* count_in = (const float*)d_in[1];
  const float* mean_in  = (const float*)d_in[2];
  const float* m2_in    = (const float*)d_in[3];
  float* y = (float*)d_out;

  const int N  = in_sizes[0];     // 16384*4096 = 2^26
  const int n4 = N >> 2;

  float* ws = (float*)d_ws;
  float* ps = ws;                  // [RED_BLOCKS]
  float* pq = ws + RED_BLOCKS;     // [RED_BLOCKS]
  float* sc = ws + 2 * RED_BLOCKS; // {new_mean, inv_std}

  vn_partial_reduce<<<RED_BLOCKS, TPB, 0, stream>>>((const v4f*)x, n4, ps, pq);
  vn_finalize<<<1, TPB, 0, stream>>>(ps, pq, count_in, mean_in, m2_in,
                                     y + (size_t)N, sc, RED_BLOCKS, (float)N);
  vn_normalize<<<NRM_BLOCKS, TPB, 0, stream>>>((const v4f*)x, (v4f*)y, sc, n4);
}